// StructurePreservationLoss_2980707303718
// MI455X (gfx1250) — compile-verified
//
#include <hip/hip_runtime.h>

typedef __attribute__((ext_vector_type(2))) float v2f;
typedef __attribute__((ext_vector_type(8))) float v8f;

#define BDIM 4
#define NDIM 2048
#define DDIM 128
#define TILE 64
#define KCHUNK 32
#define LDS_STRIDE 36   // 32 + 4 pad -> conflict-free ds_load_b64 fragment reads

// Fused Gram + structure-preservation loss.
// Each block: 64x64 tile of G_s and G_t for one batch, epilogue -> one float partial.
__global__ __launch_bounds__(256) void StructureLoss_gram_kernel(
    const float* __restrict__ S, const float* __restrict__ T,
    float* __restrict__ partial)
{
    __shared__ __attribute__((aligned(16))) float lds_si[TILE * LDS_STRIDE];
    __shared__ __attribute__((aligned(16))) float lds_ti[TILE * LDS_STRIDE];
    __shared__ __attribute__((aligned(16))) float lds_sj[TILE * LDS_STRIDE];
    __shared__ __attribute__((aligned(16))) float lds_tj[TILE * LDS_STRIDE];
    __shared__ float sq_si[TILE], sq_ti[TILE], sq_sj[TILE], sq_tj[TILE];
    __shared__ float red[256];

    const int tid  = threadIdx.x;
    const int wave = tid >> 5;
    const int lane = tid & 31;
    const int l15  = lane & 15;
    const int hl   = lane >> 4;      // half-wave select
    const int kb   = hl << 1;        // this lane's K pair base within a K=4 slice

    const int bz = blockIdx.z;
    const int i0 = blockIdx.y * TILE;
    const int j0 = blockIdx.x * TILE;

    const float* Si = S + ((size_t)bz * NDIM + i0) * DDIM;
    const float* Ti = T + ((size_t)bz * NDIM + i0) * DDIM;
    const float* Sj = S + ((size_t)bz * NDIM + j0) * DDIM;
    const float* Tj = T + ((size_t)bz * NDIM + j0) * DDIM;

    // ---- row squared norms: 256 threads cover 4 arrays x 64 rows ----
    {
        const int grp = tid >> 6;   // 0..3
        const int r   = tid & 63;
        const float* src = (grp == 0) ? Si : (grp == 1) ? Ti : (grp == 2) ? Sj : Tj;
        const float4* p = (const float4*)(src + r * DDIM);
        float acc = 0.0f;
        #pragma unroll
        for (int c = 0; c < DDIM / 4; ++c) {
            float4 q = p[c];
            acc += q.x * q.x + q.y * q.y + q.z * q.z + q.w * q.w;
        }
        float* dst = (grp == 0) ? sq_si : (grp == 1) ? sq_ti : (grp == 2) ? sq_sj : sq_tj;
        dst[r] = acc;
    }

    // wave -> subtile mapping: m subtile = wave>>1, two n subtiles at (wave&1)*32
    const int M0 = (wave >> 1) * 16;
    const int N0 = (wave & 1) * 32;
    const int N1 = N0 + 16;

    v8f cs0 = {}; v8f cs1 = {}; v8f ct0 = {}; v8f ct1 = {};

    for (int kc = 0; kc < DDIM / KCHUNK; ++kc) {
        __syncthreads();  // protect LDS reuse (also publishes sq_* on first pass)
        const int kcol = kc * KCHUNK;
        // stage 4 tiles of [64 x 32] f32: 512 float4 per tile, 2 per thread per tile
        #pragma unroll
        for (int u = 0; u < 2; ++u) {
            const int f   = tid * 2 + u;      // 0..511
            const int row = f >> 3;
            const int c4  = (f & 7) * 4;
            float4 qa = *(const float4*)(Si + row * DDIM + kcol + c4);
            float4 qb = *(const float4*)(Ti + row * DDIM + kcol + c4);
            float4 qc = *(const float4*)(Sj + row * DDIM + kcol + c4);
            float4 qd = *(const float4*)(Tj + row * DDIM + kcol + c4);
            *(float4*)&lds_si[row * LDS_STRIDE + c4] = qa;
            *(float4*)&lds_ti[row * LDS_STRIDE + c4] = qb;
            *(float4*)&lds_sj[row * LDS_STRIDE + c4] = qc;
            *(float4*)&lds_tj[row * LDS_STRIDE + c4] = qd;
        }
        __syncthreads();

        #pragma unroll
        for (int kk = 0; kk < KCHUNK / 4; ++kk) {
            const int kof = kk * 4 + kb;
            // A fragment: rows of the i-tile; B fragment (= columns of Xj^T) loads
            // identically from the j-tile rows.
            v2f a_s  = *(const v2f*)&lds_si[(M0 + l15) * LDS_STRIDE + kof];
            v2f a_t  = *(const v2f*)&lds_ti[(M0 + l15) * LDS_STRIDE + kof];
            v2f b_s0 = *(const v2f*)&lds_sj[(N0 + l15) * LDS_STRIDE + kof];
            v2f b_s1 = *(const v2f*)&lds_sj[(N1 + l15) * LDS_STRIDE + kof];
            v2f b_t0 = *(const v2f*)&lds_tj[(N0 + l15) * LDS_STRIDE + kof];
            v2f b_t1 = *(const v2f*)&lds_tj[(N1 + l15) * LDS_STRIDE + kof];
            cs0 = __builtin_amdgcn_wmma_f32_16x16x4_f32(false, a_s, false, b_s0, (short)0, cs0, false, false);
            cs1 = __builtin_amdgcn_wmma_f32_16x16x4_f32(false, a_s, false, b_s1, (short)0, cs1, false, false);
            ct0 = __builtin_amdgcn_wmma_f32_16x16x4_f32(false, a_t, false, b_t0, (short)0, ct0, false, false);
            ct1 = __builtin_amdgcn_wmma_f32_16x16x4_f32(false, a_t, false, b_t1, (short)0, ct1, false, false);
        }
    }
    __syncthreads();

    // ---- fused epilogue: pairwise-L2 + sigmoid topo, squared diffs ----
    float part = 0.0f;
    #pragma unroll
    for (int v = 0; v < 8; ++v) {
        const int mi = M0 + v + 8 * hl;          // C/D layout: VGPR v, half-wave hl
        {
            const int nj = N0 + l15;
            float gs = cs0[v], gt = ct0[v];
            float d2s = fmaxf(sq_si[mi] + sq_sj[nj] - 2.0f * gs, 0.0f);
            float d2t = fmaxf(sq_ti[mi] + sq_tj[nj] - 2.0f * gt, 0.0f);
            float rs = (d2s > 0.0f) ? sqrtf(d2s) : 0.0f;
            float rt = (d2t > 0.0f) ? sqrtf(d2t) : 0.0f;
            float dr = rs - rt;
            float ps = 1.0f / (1.0f + __expf(-gs));
            float pt = 1.0f / (1.0f + __expf(-gt));
            float dp = ps - pt;
            part += dr * dr + dp * dp;
        }
        {
            const int nj = N1 + l15;
            float gs = cs1[v], gt = ct1[v];
            float d2s = fmaxf(sq_si[mi] + sq_sj[nj] - 2.0f * gs, 0.0f);
            float d2t = fmaxf(sq_ti[mi] + sq_tj[nj] - 2.0f * gt, 0.0f);
            float rs = (d2s > 0.0f) ? sqrtf(d2s) : 0.0f;
            float rt = (d2t > 0.0f) ? sqrtf(d2t) : 0.0f;
            float dr = rs - rt;
            float ps = 1.0f / (1.0f + __expf(-gs));
            float pt = 1.0f / (1.0f + __expf(-gt));
            float dp = ps - pt;
            part += dr * dr + dp * dp;
        }
    }

    // deterministic block reduction
    red[tid] = part;
    __syncthreads();
    #pragma unroll
    for (int s = 128; s > 0; s >>= 1) {
        if (tid < s) red[tid] += red[tid + s];
        __syncthreads();
    }
    if (tid == 0) {
        const int bidx = (blockIdx.z * gridDim.y + blockIdx.y) * gridDim.x + blockIdx.x;
        partial[bidx] = red[0];
    }
}

__global__ __launch_bounds__(256) void StructureLoss_reduce_kernel(
    const float* __restrict__ partial, float* __restrict__ out, int n)
{
    __shared__ float red[256];
    float acc = 0.0f;
    for (int i = threadIdx.x; i < n; i += 256) acc += partial[i];
    red[threadIdx.x] = acc;
    __syncthreads();
    #pragma unroll
    for (int s = 128; s > 0; s >>= 1) {
        if (threadIdx.x < s) red[threadIdx.x] += red[threadIdx.x + s];
        __syncthreads();
    }
    if (threadIdx.x == 0) {
        // rel_loss + topo_loss = sum(dr^2 + dp^2) / (B*N*N)
        out[0] = red[0] * (1.0f / ((float)BDIM * (float)NDIM * (float)NDIM));
    }
}

extern "C" void kernel_launch(void* const* d_in, const int* in_sizes, int n_in,
                              void* d_out, int out_size, void* d_ws, size_t ws_size,
                              hipStream_t stream) {
    const float* S = (const float*)d_in[0];   // student_output [B,N,D] f32
    const float* T = (const float*)d_in[1];   // teacher_output [B,N,D] f32
    float* out     = (float*)d_out;           // scalar f32
    float* partial = (float*)d_ws;            // (N/64)^2 * B floats = 16 KB

    dim3 grid(NDIM / TILE, NDIM / TILE, BDIM); // 32 x 32 x 4 = 4096 blocks
    StructureLoss_gram_kernel<<<grid, 256, 0, stream>>>(S, T, partial);
    const int nblocks = (NDIM / TILE) * (NDIM / TILE) * BDIM;
    StructureLoss_reduce_kernel<<<1, 256, 0, stream>>>(partial, out, nblocks);
}